// GraphEncoder_25400436589203
// MI455X (gfx1250) — compile-verified
//
#include <hip/hip_runtime.h>

// Problem constants (match reference)
#define B_SZ   512
#define N_SZ   512
#define NNZ_SZ 8192
#define EMB    64
#define HID    100
#define HPAD   112   // 7 * 16, zero-padded hidden dim
#define NT     7     // WMMA column tiles (16 cols each)

typedef float v2f __attribute__((ext_vector_type(2)));
typedef float v8f __attribute__((ext_vector_type(8)));

// ---------------------------------------------------------------------------
// Shared WMMA inner product: one wave computes a 16xHPAD tile of X @ W.
// Xrow  : pointer to this lane's X row (row = lane & 15) with row stride baked in
// Ws    : LDS weight matrix [K][HPAD] (zero padded)
// A frag (16x4 f32): lanes 0-15 hold K={0,1}, lanes 16-31 hold K={2,3}
// B frag (4x16 f32): lanes 0-15 col=lane K={0,1}; lanes 16-31 col=lane-16 K={2,3}
// ---------------------------------------------------------------------------
template<int K>
__device__ inline void wmma_row_tile(const float* __restrict__ Xrow,
                                     const float* __restrict__ Ws,
                                     int lane, v8f acc[NT]) {
    const int khalf = (lane >> 4) << 1;   // 0 or 2
    const int col   = lane & 15;
    for (int k = 0; k < K; k += 4) {
        v2f a;
        a.x = Xrow[k + khalf];
        a.y = Xrow[k + khalf + 1];
        #pragma unroll
        for (int n = 0; n < NT; ++n) {
            const float* wp = Ws + (k + khalf) * HPAD + n * 16 + col;
            v2f bb;
            bb.x = wp[0];
            bb.y = wp[HPAD];
            acc[n] = __builtin_amdgcn_wmma_f32_16x16x4_f32(
                false, a, false, bb, (short)0, acc[n], false, false);
        }
    }
}

// C/D layout (16x16 f32): lane 0-15 -> col=lane, VGPR j -> row j;
//                         lane 16-31 -> col=lane-16, VGPR j -> row j+8
__device__ inline void store_tile(float* __restrict__ outp, int lane, const v8f acc[NT]) {
    float* p = outp + ((lane >> 4) * 8) * HPAD + (lane & 15);
    #pragma unroll
    for (int n = 0; n < NT; ++n) {
        #pragma unroll
        for (int j = 0; j < 8; ++j) {
            p[j * HPAD + n * 16] = acc[n][j];
        }
    }
}

// ---------------------------------------------------------------------------
// Kernel 1: embedding gather + X @ W1 -> support1 [B][N][HPAD]
// grid: B * (N/64), block: 128 (4 waves, each owns a 16-node tile)
// ---------------------------------------------------------------------------
__global__ __launch_bounds__(128)
void gemm1_embed(const long long* __restrict__ hyper,
                 const float* __restrict__ emb,
                 const float* __restrict__ W1,
                 float* __restrict__ support) {
    extern __shared__ float smem[];
    float* Xs = smem;                 // [64][EMB]
    float* Ws = smem + 64 * EMB;      // [EMB][HPAD], zero padded

    const int tid   = threadIdx.x;
    const int b     = blockIdx.x >> 3;
    const int node0 = (blockIdx.x & 7) * 64;

    // stage gathered embedding rows: 2 threads per row, 32 floats each
    {
        const int r    = tid >> 1;
        const int half = tid & 1;
        long long idx = hyper[(long long)b * N_SZ + node0 + r];
        const float4* src = (const float4*)(emb + idx * EMB + half * 32);
        float4* dst = (float4*)(Xs + r * EMB + half * 32);
        #pragma unroll
        for (int i = 0; i < 8; ++i) dst[i] = src[i];
    }
    // stage W1 zero-padded to [EMB][HPAD]
    for (int i = tid; i < EMB * (HPAD / 4); i += 128) {
        const int r  = i / (HPAD / 4);
        const int c4 = i % (HPAD / 4);
        float4 v = {};
        if (c4 < HID / 4) v = *(const float4*)(W1 + r * HID + c4 * 4);
        *(float4*)(Ws + r * HPAD + c4 * 4) = v;
    }
    __syncthreads();

    const int wave = tid >> 5;
    const int lane = tid & 31;
    v8f acc[NT] = {};
    wmma_row_tile<EMB>(Xs + (wave * 16 + (lane & 15)) * EMB, Ws, lane, acc);

    store_tile(support + ((long long)b * N_SZ + node0 + wave * 16) * HPAD, lane, acc);
}

// ---------------------------------------------------------------------------
// Kernel 2: X(padded) @ W2 -> support2 [B][N][HPAD]
// ---------------------------------------------------------------------------
__global__ __launch_bounds__(128)
void gemm2(const float* __restrict__ X,
           const float* __restrict__ W2,
           float* __restrict__ support) {
    extern __shared__ float smem[];
    float* Xs = smem;                  // [64][HPAD]
    float* Ws = smem + 64 * HPAD;      // [HPAD][HPAD], zero padded

    const int tid   = threadIdx.x;
    const int b     = blockIdx.x >> 3;
    const int node0 = (blockIdx.x & 7) * 64;

    for (int i = tid; i < 64 * (HPAD / 4); i += 128) {
        const int r  = i / (HPAD / 4);
        const int c4 = i % (HPAD / 4);
        *(float4*)(Xs + r * HPAD + c4 * 4) =
            *(const float4*)(X + ((long long)b * N_SZ + node0 + r) * HPAD + c4 * 4);
    }
    for (int i = tid; i < HPAD * (HPAD / 4); i += 128) {
        const int r  = i / (HPAD / 4);
        const int c4 = i % (HPAD / 4);
        float4 v = {};
        if (r < HID && c4 < HID / 4) v = *(const float4*)(W2 + r * HID + c4 * 4);
        *(float4*)(Ws + r * HPAD + c4 * 4) = v;
    }
    __syncthreads();

    const int wave = tid >> 5;
    const int lane = tid & 31;
    v8f acc[NT] = {};
    wmma_row_tile<HPAD>(Xs + (wave * 16 + (lane & 15)) * HPAD, Ws, lane, acc);

    store_tile(support + ((long long)b * N_SZ + node0 + wave * 16) * HPAD, lane, acc);
}

// ---------------------------------------------------------------------------
// Kernel 3: per-batch COO SpMM + bias.
// One workgroup per batch; accumulator [N][HPAD] fp32 lives in 229 KB LDS
// (CDNA5: 320 KB per WGP). Lanes split the hidden dim; ds_add_f32 atomics.
// FINAL=false -> write padded [B][N][HPAD]; FINAL=true -> write [B][N][HID].
// ---------------------------------------------------------------------------
template<bool FINAL>
__global__ __launch_bounds__(256)
void spmm_bias(const int* __restrict__ rows,
               const int* __restrict__ cols,
               const float* __restrict__ vals,
               const float* __restrict__ support,
               const float* __restrict__ bias,
               float* __restrict__ out) {
    extern __shared__ float acc[];    // [N_SZ][HPAD]
    const int b   = blockIdx.x;
    const int tid = threadIdx.x;

    for (int i = tid; i < N_SZ * HPAD; i += 256) acc[i] = 0.0f;
    __syncthreads();

    const int wave = tid >> 5;
    const int lane = tid & 31;
    const long long ebase = (long long)b * NNZ_SZ;
    const int e0 = wave * (NNZ_SZ / 8);
    for (int e = e0; e < e0 + (NNZ_SZ / 8); ++e) {
        const int   r = rows[ebase + e];   // all 32 lanes: same address (broadcast)
        const int   c = cols[ebase + e];
        const float v = vals[ebase + e];
        const float* sp = support + ((long long)b * N_SZ + c) * HPAD;
        float* ap = acc + r * HPAD;
        #pragma unroll
        for (int j = 0; j < 4; ++j) {
            const int h = lane + 32 * j;
            if (h < HID) atomicAdd(&ap[h], v * sp[h]);   // ds_add_f32
        }
    }
    __syncthreads();

    if (FINAL) {
        for (int i = tid; i < N_SZ * HID; i += 256) {
            const int r = i / HID;
            const int h = i % HID;
            out[((long long)b * N_SZ + r) * HID + h] = acc[r * HPAD + h] + bias[h];
        }
    } else {
        for (int i = tid; i < N_SZ * HPAD; i += 256) {
            const int h = i % HPAD;
            float v = acc[i];
            if (h < HID) v += bias[h];
            out[(long long)b * N_SZ * HPAD + i] = v;
        }
    }
}

// ---------------------------------------------------------------------------
extern "C" void kernel_launch(void* const* d_in, const int* in_sizes, int n_in,
                              void* d_out, int out_size, void* d_ws, size_t ws_size,
                              hipStream_t stream) {
    const long long* hyper    = (const long long*)d_in[0]; // int64 [B,N]
    const int*       adj_rows = (const int*)d_in[1];       // int32 [B,NNZ]
    const int*       adj_cols = (const int*)d_in[2];
    const float*     adj_vals = (const float*)d_in[3];
    const float*     emb      = (const float*)d_in[4];     // [ENTITY, EMB]
    const float*     W1       = (const float*)d_in[5];     // [EMB, HID]
    const float*     b1       = (const float*)d_in[6];
    const float*     W2       = (const float*)d_in[7];     // [HID, HID]
    const float*     b2       = (const float*)d_in[8];
    float* out = (float*)d_out;

    float* buf0 = (float*)d_ws;                                // support (padded)
    float* buf1 = buf0 + (size_t)B_SZ * N_SZ * HPAD;           // x1 (padded)

    const size_t smem1 = (size_t)(64 * EMB + EMB * HPAD) * sizeof(float);    // 45 KB
    const size_t smem2 = (size_t)(64 * HPAD + HPAD * HPAD) * sizeof(float);  // 79 KB
    const size_t smem3 = (size_t)N_SZ * HPAD * sizeof(float);                // 229 KB

    // opt into >64KB dynamic LDS (CDNA5 allows up to 320KB per workgroup)
    hipFuncSetAttribute(reinterpret_cast<const void*>(&gemm2),
                        hipFuncAttributeMaxDynamicSharedMemorySize, (int)smem2);
    hipFuncSetAttribute(reinterpret_cast<const void*>(&spmm_bias<false>),
                        hipFuncAttributeMaxDynamicSharedMemorySize, (int)smem3);
    hipFuncSetAttribute(reinterpret_cast<const void*>(&spmm_bias<true>),
                        hipFuncAttributeMaxDynamicSharedMemorySize, (int)smem3);

    const int gemm_grid = B_SZ * (N_SZ / 64);

    // layer 1
    gemm1_embed<<<gemm_grid, 128, smem1, stream>>>(hyper, emb, W1, buf0);
    spmm_bias<false><<<B_SZ, 256, smem3, stream>>>(adj_rows, adj_cols, adj_vals,
                                                   buf0, b1, buf1);
    // layer 2
    gemm2<<<gemm_grid, 128, smem2, stream>>>(buf1, W2, buf0);
    spmm_bias<true><<<B_SZ, 256, smem3, stream>>>(adj_rows, adj_cols, adj_vals,
                                                  buf0, b2, out);
}